// Regress_hyper_parameters_Model_14697378087176
// MI455X (gfx1250) — compile-verified
//
#include <hip/hip_runtime.h>

#define NV      64
#define NP      2016          // 64*63/2 pairs
#define NPF     (NP * 3)      // 6048 floats of pair data per batch row
#define THREADS 256

// ---- CDNA5 async global->LDS path (guarded; falls back to plain b128) ----
#if defined(__HIP_DEVICE_COMPILE__) && defined(__gfx1250__)
#  if __has_builtin(__builtin_amdgcn_global_load_async_to_lds_b128)
#    define HAVE_ASYNC_B128 1
#  endif
#  if __has_builtin(__builtin_amdgcn_s_wait_asynccnt)
#    define HAVE_WAIT_ASYNC 1
#  endif
#endif

// Exact pointee type the builtin expects (per hipcc diagnostic):
//   "__attribute__((__vector_size__(4 * sizeof(int)))) int *"
typedef int v4i __attribute__((vector_size(4 * sizeof(int))));
#define AS_GLOBAL __attribute__((address_space(1)))
#define AS_LDS    __attribute__((address_space(3)))

__device__ __forceinline__ void stage_b128(const float* g, float* l) {
#if defined(HAVE_ASYNC_B128)
  __builtin_amdgcn_global_load_async_to_lds_b128(
      (AS_GLOBAL v4i*)g,   // global source (C-style cast: drops const + addrspacecast)
      (AS_LDS    v4i*)l,   // LDS destination
      /*offset=*/0, /*cpol=*/0);
#else
  *(float4*)l = *(const float4*)g;
#endif
}

__device__ __forceinline__ void wait_stage_done() {
#if defined(HAVE_ASYNC_B128)
#  if defined(HAVE_WAIT_ASYNC)
  __builtin_amdgcn_s_wait_asynccnt(0);
#  else
  asm volatile("s_wait_asynccnt 0" ::: "memory");
#  endif
#endif
}

// start of triu row i in flat pair index space: i*(127-i)/2 (always integral)
__device__ __forceinline__ int row_start(int i) { return (i * (127 - i)) >> 1; }

__global__ __launch_bounds__(THREADS)
void fused_pair_score_kernel(const float* __restrict__ views,   // [B,64,7]
                             const float* __restrict__ pairs,   // [B,2016,3]
                             const float* __restrict__ a1p,
                             const float* __restrict__ a2p,
                             const float* __restrict__ gp,
                             const float* __restrict__ sp,
                             const float* __restrict__ npv,
                             const float* __restrict__ dp,
                             float* __restrict__ out)
{
  __shared__ __align__(16) float pbuf[NPF];   // 24,192 B: one row of pair data
  __shared__ float w[NV];                     // per-view fused weights
  __shared__ float red[THREADS / 32];

  const int tid = threadIdx.x;
  const int b   = blockIdx.x;

  // ---- kick off async staging of this row's pair data into LDS ----
  const float* grow = pairs + (size_t)b * NPF;
  #pragma unroll
  for (int pass = 0; pass < 6; ++pass) {
    int dw = (pass * THREADS + tid) * 4;      // 1512 b128 chunks total
    if (dw < NPF) stage_b128(grow + dw, &pbuf[dw]);
  }

  // ---- uniform scalars; fold log2(e) so exp() is a single v_exp_f32 ----
  const float LOG2E = 1.4426950408889634f;
  const float a1 = *a1p, a2 = *a2p, g = *gp, s = *sp, n = *npv, d = *dp;
  const float k_a1 = LOG2E / (2.0f * a1 * a1);
  const float k_a2 = LOG2E / (2.0f * a2 * a2);
  const float k_g  = LOG2E / (2.0f * g  * g );
  const float k_s  = LOG2E / (2.0f * s  * s );
  const float k_n  = LOG2E / (2.0f * n  * n );
  const float k_d  = LOG2E / (2.0f * d  * d );

  // ---- per-view fused weight: w = mask * exp(-(x4^4/2sg^2 + x5^2/2sn^2 + x6^2/2sd^2)) ----
  if (tid < NV) {
    const float* vr = views + ((size_t)b * NV + tid) * 7;
    float x0 = vr[0], x4 = vr[4], x5 = vr[5], x6 = vr[6];
    float x4sq = x4 * x4;
    float t = x4sq * x4sq * k_g + x5 * x5 * k_n + x6 * x6 * k_d;
    float m = (x0 == 1.0f) ? 1.0f : 0.0f;
    w[tid] = m * __builtin_amdgcn_exp2f(-t);
  }

  wait_stage_done();
  __syncthreads();

  // ---- pair loop: 8 strided iterations per thread, register accumulate ----
  const float ATH = 20.0f / 90.0f;
  float acc = 0.0f;
  for (int p = tid; p < NP; p += THREADS) {
    float c0 = pbuf[p * 3 + 0];
    float a  = pbuf[p * 3 + 1];
    float c2 = pbuf[p * 3 + 2];

    // flat triu index -> (i, j), closed form + exact integer correction
    int i = (int)(0.5f * (127.0f - sqrtf(16129.0f - 8.0f * (float)p)));
    i = (i < 0) ? 0 : ((i > 62) ? 62 : i);
    while (i < 62 && row_start(i + 1) <= p) ++i;
    while (i > 0  && row_start(i) > p)      --i;
    int j = p - row_start(i) + i + 1;

    float pm  = (c0 != 0.0f) ? 1.0f : 0.0f;
    float val = 0.0f;                               // a == ATH -> exp(-inf) = 0
    if (a != ATH) {
      float ka = (a < ATH) ? k_a1 : k_a2;
      val = pm * __builtin_amdgcn_exp2f(-(a * a * ka + c2 * c2 * k_s));
    }
    acc += val * w[i] * w[j];
  }

  // ---- wave32 shuffle reduction, then cross-wave via LDS ----
  #pragma unroll
  for (int off = 16; off > 0; off >>= 1)
    acc += __shfl_down(acc, off, 32);
  if ((tid & 31) == 0) red[tid >> 5] = acc;
  __syncthreads();
  if (tid == 0) {
    float sres = red[0];
    #pragma unroll
    for (int k = 1; k < THREADS / 32; ++k) sres += red[k];
    out[b] = sres;
  }
}

extern "C" void kernel_launch(void* const* d_in, const int* in_sizes, int n_in,
                              void* d_out, int out_size, void* d_ws, size_t ws_size,
                              hipStream_t stream) {
  // setup_inputs order:
  // 0: views [B,64,7] f32      1: view_pairs [B,2016,3] f32
  // 2: point_attribute [B,2]   (unused by the reference math)
  // 3: alpha_sigma1  4: alpha_sigma2  5: gsd_sigma
  // 6: scale_sigma   7: normal_sigma  8: distortion_sigma
  const float* views = (const float*)d_in[0];
  const float* pairs = (const float*)d_in[1];
  const int B = in_sizes[0] / (NV * 7);

  fused_pair_score_kernel<<<B, THREADS, 0, stream>>>(
      views, pairs,
      (const float*)d_in[3], (const float*)d_in[4], (const float*)d_in[5],
      (const float*)d_in[6], (const float*)d_in[7], (const float*)d_in[8],
      (float*)d_out);
}